// SimplestSpline_69724499083956
// MI455X (gfx1250) — compile-verified
//
#include <hip/hip_runtime.h>

// ---------------------------------------------------------------------------
// Piecewise-linear per-(b,c) spline LUT over a (8,3,1024,1024) fp32 image.
// Pure streaming problem: ~200MB traffic, ~0.2 GFLOP -> HBM-bound (~8.6us at
// 23.3 TB/s). Strategy: per-plane 17-entry (y_right, slope) LUT in LDS,
// double-buffered async global->LDS staging (CDNA5 ASYNCcnt path) for the
// input stream, nontemporal B128 stores for the output stream.
// ---------------------------------------------------------------------------

typedef float v4f __attribute__((ext_vector_type(4)));
typedef int   v4i __attribute__((vector_size(4 * sizeof(int))));

#define N_KNOTS 16
#define N_SEG   17          // N_KNOTS + 1 linear segments on [0,1]

constexpr int TPB              = 256;            // 8 wave32 per block
constexpr int ITER             = 16;             // float4 chunks per thread
constexpr int HW               = 1024 * 1024;    // pixels per plane
constexpr int PLANES           = 8 * 3;          // B*C independent curves
constexpr int F4_PER_PLANE     = HW / 4;                      // 262144
constexpr int F4_PER_BLOCK     = TPB * ITER;                  // 4096
constexpr int BLOCKS_PER_PLANE = F4_PER_PLANE / F4_PER_BLOCK; // 64

#if defined(__gfx1250__) &&                                          \
    __has_builtin(__builtin_amdgcn_global_load_async_to_lds_b128) && \
    __has_builtin(__builtin_amdgcn_s_wait_asynccnt)
#define USE_ASYNC_LDS 1
#else
#define USE_ASYNC_LDS 0
#endif

__device__ __forceinline__ float eval_spline(float x, const float2* __restrict__ tab) {
  // knots uniform: segment index = clamp(floor(x*17), 0, 16)
  int idx = (int)(x * (float)N_SEG);          // trunc == floor for x >= 0
  idx = idx < 0 ? 0 : (idx > N_SEG - 1 ? N_SEG - 1 : idx);
  float2 e = tab[idx];                        // ds_load_b64: {y_right, slope}
  float xs_right = (float)(idx + 1) * (1.0f / (float)N_SEG);
  // out = y_right - (xs_right - x)*slope == fma(x - xs_right, slope, y_right)
  return __builtin_fmaf(x - xs_right, e.y, e.x);
}

__global__ __launch_bounds__(TPB) void spline_lut_kernel(
    const float* __restrict__ raw, const float* __restrict__ params,
    float* __restrict__ out) {
  __shared__ float2 tab[N_SEG];        // per-segment (y_right, slope)
  __shared__ v4f    stage[2][TPB];     // 8KB double-buffer for async staging

  const int tid   = threadIdx.x;
  const int plane = blockIdx.y;        // b*C + c : one curve per plane

  // Build the LUT once per block. ys_pad = [0, params[plane][0..15], 1].
  if (tid < N_SEG) {
    float y_lo = (tid == 0)         ? 0.0f : params[plane * N_KNOTS + tid - 1];
    float y_hi = (tid == N_SEG - 1) ? 1.0f : params[plane * N_KNOTS + tid];
    tab[tid] = make_float2(y_hi, (y_hi - y_lo) * (float)N_SEG); // slope = dy/dx, dx=1/17
  }
  __syncthreads();

  const size_t base = (size_t)plane * F4_PER_PLANE + (size_t)blockIdx.x * F4_PER_BLOCK;
  const v4f* __restrict__ src = (const v4f*)raw + base;
  v4f* __restrict__ dst       = (v4f*)out + base;

#if USE_ASYNC_LDS
  typedef __attribute__((address_space(3))) void* lvoid_t;
  typedef __attribute__((address_space(3))) v4i*  lv4i_t;

  // Prime the pipeline: chunk 0 -> stage[0].
  __builtin_amdgcn_global_load_async_to_lds_b128(
      (v4i*)(src + tid), (lv4i_t)(lvoid_t)(void*)&stage[0][tid], 0, 0);

#pragma unroll
  for (int i = 0; i < ITER; ++i) {
    if (i + 1 < ITER) {
      // Prefetch next chunk into the other buffer, then wait for the
      // oldest async op only (in-order completion => current chunk landed).
      __builtin_amdgcn_global_load_async_to_lds_b128(
          (v4i*)(src + (i + 1) * TPB + tid),
          (lv4i_t)(lvoid_t)(void*)&stage[(i + 1) & 1][tid], 0, 0);
      __builtin_amdgcn_s_wait_asynccnt(1);
    } else {
      __builtin_amdgcn_s_wait_asynccnt(0);
    }
    v4f v = stage[i & 1][tid];          // each lane reads back its own slot
    v4f r;
    r.x = eval_spline(v.x, tab);
    r.y = eval_spline(v.y, tab);
    r.z = eval_spline(v.z, tab);
    r.w = eval_spline(v.w, tab);
    __builtin_nontemporal_store(r, dst + i * TPB + tid);  // TH=NT store
  }
#else
  // Fallback: straight B128 streaming loads (still saturates HBM with
  // enough waves in flight).
#pragma unroll 4
  for (int i = 0; i < ITER; ++i) {
    v4f v = __builtin_nontemporal_load(src + i * TPB + tid);
    v4f r;
    r.x = eval_spline(v.x, tab);
    r.y = eval_spline(v.y, tab);
    r.z = eval_spline(v.z, tab);
    r.w = eval_spline(v.w, tab);
    __builtin_nontemporal_store(r, dst + i * TPB + tid);
  }
#endif
}

extern "C" void kernel_launch(void* const* d_in, const int* in_sizes, int n_in,
                              void* d_out, int out_size, void* d_ws, size_t ws_size,
                              hipStream_t stream) {
  const float* raw    = (const float*)d_in[0];   // (8,3,1024,1024) fp32
  const float* params = (const float*)d_in[1];   // (8, 48) fp32
  float* out          = (float*)d_out;           // (8,3,1024,1024) fp32

  dim3 grid(BLOCKS_PER_PLANE, PLANES);           // (64, 24)
  spline_lut_kernel<<<grid, TPB, 0, stream>>>(raw, params, out);
}